// D2IM_Net_15444702396680
// MI455X (gfx1250) — compile-verified
//
#include <hip/hip_runtime.h>

#define IMG 224
#define IMG2 (IMG * IMG)      // 50176
#define BATCH 32
#define NPTS 32768
#define NBLOCKS 1024          // 32 images * 32 chunks of 1024 points
#define NACC 7

typedef __attribute__((ext_vector_type(2))) float v2f;
typedef __attribute__((ext_vector_type(8))) float v8f;

__device__ __forceinline__ float wave_sum(float v) {
    v += __shfl_xor(v, 16, 32);
    v += __shfl_xor(v, 8, 32);
    v += __shfl_xor(v, 4, 32);
    v += __shfl_xor(v, 2, 32);
    v += __shfl_xor(v, 1, 32);
    return v;
}

__global__ __launch_bounds__(256) void d2im_main(
    const float* __restrict__ gt_points,
    const float* __restrict__ gt_values,
    const float* __restrict__ gt_gradients,
    const float* __restrict__ mc_image,
    const float* __restrict__ gt_transmat,
    const float* __restrict__ scale,
    const float* __restrict__ base_values,
    const float* __restrict__ pred_disp,
    float* __restrict__ partials)
{
    const int t    = threadIdx.x;
    const int lane = t & 31;
    const int w    = t >> 5;
    const int blk  = blockIdx.x;
    const int b    = blk >> 5;          // image index
    const int base = (blk & 31) * 1024; // first point of this block's chunk

    const float scl = scale[0];
    const float* T = gt_transmat + b * 12;     // T[r][c] at T[r*3+c], shape (4,3)
    const float T02 = T[2], T12 = T[5], T22 = T[8];

    // A matrix = transmat^T as 16x4 f32 (rows 0..2 valid, rest zero), built branchlessly.
    // Layout: lanes 0-15 -> M=lane, VGPR0=K0, VGPR1=K1; lanes 16-31 -> VGPR0=K2, VGPR1=K3.
    const int  m  = lane & 15;
    const bool hi = lane >= 16;
    {
        // computed below into A
    }
    const int  mm    = (m < 3) ? m : 0;      // clamp so address is always valid
    const bool valid = (m < 3);
    const float ta0  = T[(hi ? 6 : 0) + mm]; // A[m][k] = T[k][m]; K = hi?2:0
    const float ta1  = T[(hi ? 9 : 3) + mm]; //                    K = hi?3:1
    v2f A;
    A.x = valid ? ta0 : 0.f;
    A.y = valid ? ta1 : 0.f;

    const float* PF = pred_disp + (size_t)b * 2 * IMG2;        // front channel
    const float* PB = PF + IMG2;                               // back channel
    const float* G4 = mc_image + ((size_t)b * 6 + 4) * IMG2;   // mc ch4 (gt_normal y)
    const float* G5 = G4 + IMG2;                               // mc ch5 (gt_normal z)

    float acc0 = 0.f, acc1 = 0.f, acc2 = 0.f, acc3 = 0.f, acc4 = 0.f, acc5 = 0.f, acc6 = 0.f;

    for (int it = 0; it < 4; ++it) {
        const int pbase = base + it * 256 + w * 32;   // this wave's 32 points
        const float* P = gt_points + ((size_t)b * NPTS + pbase) * 3;

        // B matrices (4x16 f32): column n = point n's homogeneous coords.
        // lanes 0-15: VGPR0=x, VGPR1=y ; lanes 16-31: VGPR0=z, VGPR1=1
        // All loads unconditional (P[3m+1] is in-bounds for every lane); selects via cndmask.
        const int off = 3 * m + (hi ? 2 : 0);
        const float p0 = P[off];            // x (lo lanes) or z (hi lanes), point m
        const float p1 = P[3 * m + 1];      // y of point m
        const float q0 = P[off + 48];       // same for point m+16
        const float q1 = P[3 * m + 49];
        v2f B1, B2;
        B1.x = p0;  B1.y = hi ? 1.0f : p1;
        B2.x = q0;  B2.y = hi ? 1.0f : q1;

        // D[m][n] = sum_k T[k][m] * p_n[k]  -> rows 0..2 = homo x,y,z of point n.
        // D layout: VGPR r, lanes 0-15 hold D[r][lane] -> lanes 0-15, VGPRs 0..2 = (hx,hy,hz).
        v8f Cz = {};
        v8f d1 = __builtin_amdgcn_wmma_f32_16x16x4_f32(false, A, false, B1, (short)0, Cz, false, false);
        v8f d2 = __builtin_amdgcn_wmma_f32_16x16x4_f32(false, A, false, B2, (short)0, Cz, false, false);

        // Redistribute at FULL exec: shfl_xor(16) swaps half-waves, so hi lane l gets
        // d2's lane l-16 value (= homo of point pbase+l). Straight-line, no divergence.
        const float e0 = __shfl_xor(d2[0], 16, 32);
        const float e1 = __shfl_xor(d2[1], 16, 32);
        const float e2 = __shfl_xor(d2[2], 16, 32);
        const float hx = hi ? e0 : d1[0];
        const float hy = hi ? e1 : d1[1];
        const float hz = hi ? e2 : d1[2];

        const size_t gp = (size_t)b * NPTS + (pbase + lane);

        const float z  = hz;
        const float px = hx / z;
        const float py = hy / z;
        const int ix = (int)fminf(fmaxf(px, 0.f), 223.f);  // trunc-toward-zero + clip == ref
        const int iy = (int)fminf(fmaxf(py, 0.f), 223.f);
        const int idx = iy * IMG + ix;

        const float gx = gt_gradients[gp * 3 + 0];
        const float gy = gt_gradients[gp * 3 + 1];
        const float gz = gt_gradients[gp * 3 + 2];
        const float pgz = gx * T02 + gy * T12 + gz * T22;
        const float sgn = (pgz > 0.f) ? 1.f : ((pgz < 0.f) ? -1.f : 0.f);
        const float fw  = 0.5f - 0.5f * sgn;
        const float bw  = 1.f - fw;

        const float gv = gt_values[gp];
        const float bv = base_values[gp];
        const float db = bv - gv;
        acc0 += db * db;

        const float pdf = PF[idx];
        const float pdb = PB[idx];
        acc1 += fw;
        acc2 += fw * fabsf(bv + pdf - gv);
        acc3 += bw;
        acc4 += bw * fabsf(bv + pdb - gv);

        // Branchless boundary handling: clamp address (cndmask), mask value (cndmask).
        const bool bx1 = (ix >= 1), bx2 = (ix >= 2);
        const bool by1 = (iy <= 222), by2 = (iy <= 221);
        const float lx1 = PF[bx1 ? idx - 1 : idx];
        const float lx2 = PF[bx2 ? idx - 2 : idx];
        const float ly1 = PF[by1 ? idx + IMG     : idx];
        const float ly2 = PF[by2 ? idx + 2 * IMG : idx];
        const float pxm1 = bx1 ? lx1 : 0.f;
        const float pxm2 = bx2 ? lx2 : 0.f;
        const float pyp1 = by1 ? ly1 : 0.f;
        const float pyp2 = by2 ? ly2 : 0.f;
        // pred_n2 = 0.5*((in - 2*in[x-1] + in[x-2]) + (in - 2*in[y+1] + in[y+2]))
        const float pn2 = 0.5f * ((pdf - 2.f * pxm1 + pxm2) + (pdf - 2.f * pyp1 + pyp2));

        // gt_n2 = 0.5*((g5 - g5[x-1]) + (g4 - g4[y+1]))
        const float g5v = G5[idx];
        const float g5l = G5[bx1 ? idx - 1 : idx];
        const float g4v = G4[idx];
        const float g4l = G4[by1 ? idx + IMG : idx];
        const float g5m = bx1 ? g5l : 0.f;
        const float g4p = by1 ? g4l : 0.f;
        const float gn2 = 0.5f * ((g5v - g5m) + (g4v - g4p));

        const float ppn2 = pn2 * 49.f * scl / (2.f * z);
        const float lw   = (gv < 0.1f) ? fw : 0.f;
        const float dsum = ppn2 + gn2;
        acc5 += lw;
        acc6 += lw * dsum * dsum;
    }

    // Block reduction: wave32 shfl-xor, then LDS across 8 waves.
    __shared__ float red[8][NACC];
    float s[NACC] = {acc0, acc1, acc2, acc3, acc4, acc5, acc6};
#pragma unroll
    for (int k = 0; k < NACC; ++k) {
        float v = wave_sum(s[k]);
        if (lane == 0) red[w][k] = v;
    }
    __syncthreads();
    if (t == 0) {
#pragma unroll
        for (int k = 0; k < NACC; ++k) {
            float v = 0.f;
            for (int ww = 0; ww < 8; ++ww) v += red[ww][k];
            partials[(size_t)blk * NACC + k] = v;
        }
    }
}

__global__ __launch_bounds__(256) void d2im_finalize(
    const float* __restrict__ partials, float* __restrict__ out)
{
    __shared__ double sh[NACC][256];
    const int t = threadIdx.x;
    double a[NACC] = {0, 0, 0, 0, 0, 0, 0};
    for (int i = t; i < NBLOCKS; i += 256)
#pragma unroll
        for (int k = 0; k < NACC; ++k)
            a[k] += (double)partials[(size_t)i * NACC + k];
#pragma unroll
    for (int k = 0; k < NACC; ++k) sh[k][t] = a[k];
    __syncthreads();
    for (int srd = 128; srd > 0; srd >>= 1) {
        if (t < srd)
#pragma unroll
            for (int k = 0; k < NACC; ++k) sh[k][t] += sh[k][t + srd];
        __syncthreads();
    }
    if (t == 0) {
        const double sb  = sh[0][0];
        const double sfw = sh[1][0], sf   = sh[2][0];
        const double sbw = sh[3][0], sbk  = sh[4][0];
        const double slw = sh[5][0], slap = sh[6][0];
        out[0] = (float)(sb / (double)((double)BATCH * (double)NPTS));
        out[1] = (float)(0.5 * (sf / sfw + sbk / sbw));
        out[2] = (float)(slap / slw);
    }
}

extern "C" void kernel_launch(void* const* d_in, const int* in_sizes, int n_in,
                              void* d_out, int out_size, void* d_ws, size_t ws_size,
                              hipStream_t stream) {
    (void)in_sizes; (void)n_in; (void)out_size; (void)ws_size;
    const float* gt_points    = (const float*)d_in[0];
    const float* gt_values    = (const float*)d_in[1];
    const float* gt_gradients = (const float*)d_in[2];
    const float* mc_image     = (const float*)d_in[3];
    const float* gt_transmat  = (const float*)d_in[4];
    const float* scale        = (const float*)d_in[5];
    const float* base_values  = (const float*)d_in[6];
    const float* pred_disp    = (const float*)d_in[7];
    float* out      = (float*)d_out;
    float* partials = (float*)d_ws;   // NBLOCKS*NACC floats = 28 KB, fully overwritten each call

    d2im_main<<<NBLOCKS, 256, 0, stream>>>(gt_points, gt_values, gt_gradients, mc_image,
                                           gt_transmat, scale, base_values, pred_disp, partials);
    d2im_finalize<<<1, 256, 0, stream>>>(partials, out);
}